// Ours_19430432047507
// MI455X (gfx1250) — compile-verified
//
#include <hip/hip_runtime.h>
#include <hip/hip_bf16.h>
#include <math.h>

// ---------------------------------------------------------------------------
// MI455X (gfx1250) implementation.
// Roofline: sim output = 576 MB -> ~25us @ 23.3 TB/s (store bound); score
// output = 49 MB. e_all = 3 MB -> fully L2 resident (192 MB L2). Both big
// K=64 GEMMs use V_WMMA_F32_16X16X4_F32 (fp32-faithful matrix path, wave32).
// Streaming outputs use non-temporal stores so L2 stays reserved for the
// GEMM operands.
// ---------------------------------------------------------------------------

#define NBI   12000
#define NBU   100000
#define DD    64
#define BB    1024
#define LL    20
#define BKK   10
#define NOLD  20

typedef float v2f __attribute__((ext_vector_type(2)));
typedef float v8f __attribute__((ext_vector_type(8)));

__device__ __forceinline__ float sigf(float x) { return 1.0f / (1.0f + __expf(-x)); }

// ---------------------------------------------------------------------------
// Kernel 1: fused front end. One 64-thread block per (n,b) pair (4096 blocks).
// Gathers baskets into LDS, computes k=relu(basket@Wk+bk), attention over L,
// l_vec, the 3 EMA scans, gate, final_each. Emits s1,s2,s3,l_vec,final_each.
// ---------------------------------------------------------------------------
__global__ __launch_bounds__(64) void front_kernel(
    const int* __restrict__ click, const int* __restrict__ favor,
    const int* __restrict__ cart,  const int* __restrict__ buy,
    const int* __restrict__ userData,
    const float* __restrict__ item_emb, const float* __restrict__ user_emb,
    const float* __restrict__ user_wd,  const float* __restrict__ alpha,
    const float* __restrict__ gate_w,   const float* __restrict__ gate_b,
    const float* __restrict__ Wk,       const float* __restrict__ bk,
    float* __restrict__ s1o, float* __restrict__ s2o, float* __restrict__ s3o,
    float* __restrict__ lvo, float* __restrict__ feo)
{
    __shared__ float sb[LL][DD];     // basket rows
    __shared__ float ksh[LL][DD];    // relu(k)
    __shared__ float eu[DD];
    __shared__ float red[DD];
    __shared__ float attn_s[LL];
    __shared__ float sc_s[LL];
    __shared__ float g_s;

    const int nb = blockIdx.x;
    const int n = nb >> 10, b = nb & 1023;
    const int t = threadIdx.x;                 // 0..63 : d (or e) index
    const int* idx = (n == 0 ? click : n == 1 ? favor : n == 2 ? cart : buy)
                     + b * LL * BKK;
    const int u = userData[b];

    // gather baskets: sb[l][t] = sum_j item_emb[idx[l][j], t]
    for (int l = 0; l < LL; ++l) {
        float acc = 0.0f;
        #pragma unroll
        for (int j = 0; j < BKK; ++j) {
            int it = idx[l * BKK + j];
            acc += item_emb[it * DD + t];
        }
        sb[l][t] = acc;
    }
    eu[t] = user_emb[u * DD + t];
    __syncthreads();

    // k[l][e=t] = relu( sum_d sb[l][d] * Wk[n][d][e] + bk[n][e] )
    float kacc[LL];
    #pragma unroll
    for (int l = 0; l < LL; ++l) kacc[l] = bk[n * DD + t];
    for (int d = 0; d < DD; ++d) {
        float wv = Wk[(n * DD + d) * DD + t];
        #pragma unroll
        for (int l = 0; l < LL; ++l) kacc[l] = fmaf(sb[l][d], wv, kacc[l]);
    }
    #pragma unroll
    for (int l = 0; l < LL; ++l) ksh[l][t] = fmaxf(kacc[l], 0.0f);
    __syncthreads();

    // scores (masked) -- threads 0..19 own one l each
    if (t < LL) {
        float s = 0.0f, rs = 0.0f;
        for (int e = 0; e < DD; ++e) { s += ksh[t][e] * eu[e]; rs += sb[t][e]; }
        sc_s[t] = (rs != 0.0f) ? s * 0.125f : -1.0e9f;   // scale = 1/sqrt(64)
    }
    __syncthreads();
    if (t == 0) {
        float m = sc_s[0];
        for (int l = 1; l < LL; ++l) m = fmaxf(m, sc_s[l]);
        float ssum = 0.0f;
        for (int l = 0; l < LL; ++l) { float e = __expf(sc_s[l] - m); attn_s[l] = e; ssum += e; }
        float inv = 1.0f / ssum;
        for (int l = 0; l < LL; ++l) attn_s[l] *= inv;
    }
    __syncthreads();

    // l_vec + EMA scans, per d = t
    float lv = 0.0f;
    #pragma unroll
    for (int l = 0; l < LL; ++l) lv = fmaf(attn_s[l], sb[l][t], lv);

    float w1 = sigf(user_wd[u * 8 + 2 * n]);
    float w2 = sigf(user_wd[u * 8 + 2 * n + 1]);
    float coef = 1.0f - w2 * w1;
    float h1 = sb[0][t];
    #pragma unroll
    for (int l = 1; l < LL; ++l) h1 = w1 * h1 + coef * sb[l][t];
    float h2 = sb[LL / 2][t];
    #pragma unroll
    for (int l = LL / 2 + 1; l < LL; ++l) h2 = w1 * h2 + coef * sb[l][t];
    float h3 = w1 * sb[LL - 2][t] + coef * sb[LL - 1][t];

    float a0 = alpha[0], a1 = alpha[1];
    float fn = h1 + a0 * (h1 - h2) + a1 * (h1 - h3);

    red[t] = fn * gate_w[t] + lv * gate_w[DD + t];
    __syncthreads();
    if (t == 0) {
        float s = gate_b[0];
        for (int i = 0; i < DD; ++i) s += red[i];
        g_s = sigf(s);
    }
    __syncthreads();
    float g = g_s;

    int o = nb * DD + t;
    s1o[o] = h1; s2o[o] = h2; s3o[o] = h3; lvo[o] = lv;
    feo[o] = g * fn + (1.0f - g) * lv;
}

// ---------------------------------------------------------------------------
// Kernel 2: w_mix mixing + gate -> final (B,64). One 64-thread block per b.
// ---------------------------------------------------------------------------
__global__ __launch_bounds__(64) void mix_kernel(
    const float* __restrict__ s1o, const float* __restrict__ s2o,
    const float* __restrict__ s3o, const float* __restrict__ lvo,
    const float* __restrict__ w_mix, const float* __restrict__ alpha,
    const float* __restrict__ gate_w, const float* __restrict__ gate_b,
    float* __restrict__ fino)
{
    __shared__ float red[DD];
    __shared__ float g_s;
    const int b = blockIdx.x, t = threadIdx.x;

    float wm[4][4];
    #pragma unroll
    for (int r = 0; r < 4; ++r) {
        float v0 = w_mix[r * 4 + 0], v1 = w_mix[r * 4 + 1];
        float v2 = w_mix[r * 4 + 2], v3 = w_mix[r * 4 + 3];
        float m = fmaxf(fmaxf(v0, v1), fmaxf(v2, v3));
        float e0 = __expf(v0 - m), e1 = __expf(v1 - m);
        float e2 = __expf(v2 - m), e3 = __expf(v3 - m);
        float inv = 1.0f / (e0 + e1 + e2 + e3);
        wm[r][0] = e0 * inv; wm[r][1] = e1 * inv; wm[r][2] = e2 * inv; wm[r][3] = e3 * inv;
    }

    float s1m = 0.f, s2m = 0.f, s3m = 0.f, lm = 0.f;
    #pragma unroll
    for (int n = 0; n < 4; ++n) {
        int o = (n * BB + b) * DD + t;
        s1m += wm[0][n] * s1o[o];
        s2m += wm[1][n] * s2o[o];
        s3m += wm[2][n] * s3o[o];
        lm  += wm[3][n] * lvo[o];
    }
    float a0 = alpha[0], a1 = alpha[1];
    float f = s1m + a0 * (s1m - s2m) + a1 * (s1m - s3m);

    red[t] = f * gate_w[t] + lm * gate_w[DD + t];
    __syncthreads();
    if (t == 0) {
        float s = gate_b[0];
        for (int i = 0; i < DD; ++i) s += red[i];
        g_s = sigf(s);
    }
    __syncthreads();
    float g = g_s;
    fino[b * DD + t] = g * f + (1.0f - g) * lm;
}

// ---------------------------------------------------------------------------
// Kernel 3: dense score GEMM with V_WMMA_F32_16X16X4_F32.
// score[b,e] = final[b].e_all[e] + e_user[b].u_w[:,e] + u_b[e]   (oh-free base)
// Each wave computes a 16(M)x64(N) strip: 4 v8f accumulators, K=128 fused.
// ---------------------------------------------------------------------------
__global__ __launch_bounds__(256) void score_gemm(
    const float* __restrict__ fin,       // (1024,64)
    const float* __restrict__ user_emb,  // (100000,64)
    const int*   __restrict__ userData,  // (1024,)
    const float* __restrict__ item_emb,  // e_all = item_emb + 64
    const float* __restrict__ u_w,       // (64,12000) row-major over k
    const float* __restrict__ u_b,       // (12000,)
    float* __restrict__ score)           // (1024,12000)
{
    const float* e_all = item_emb + DD;
    const int wid  = blockIdx.x * 8 + (threadIdx.x >> 5);
    const int lane = threadIdx.x & 31;
    const int l16  = lane & 15;
    const int half = lane >> 4;
    const int mTile  = wid & 63;       // 64 M tiles (1024 rows)
    const int nSuper = wid >> 6;       // 188 N super-tiles (64 cols each)
    const int mBase = mTile * 16;
    const int nBase = nSuper * 64;
    const int arow  = mBase + l16;
    const int uidx  = userData[arow];

    const v8f vzero = {0.f, 0.f, 0.f, 0.f, 0.f, 0.f, 0.f, 0.f};
    v8f acc[4] = {vzero, vzero, vzero, vzero};

    int ccol[4];
    #pragma unroll
    for (int q = 0; q < 4; ++q) {
        int col = nBase + q * 16 + l16;
        ccol[q] = col < NBI ? col : NBI - 1;   // clamp (keep EXEC all-1s)
    }

    // Phase 1: final @ e_all^T  (K = 0..63)
    #pragma unroll
    for (int k = 0; k < DD; k += 4) {
        v2f a = *(const v2f*)(fin + arow * DD + k + 2 * half);
        #pragma unroll
        for (int q = 0; q < 4; ++q) {
            v2f bmat = *(const v2f*)(e_all + ccol[q] * DD + k + 2 * half);
            acc[q] = __builtin_amdgcn_wmma_f32_16x16x4_f32(
                false, a, false, bmat, (short)0, acc[q], false, false);
        }
    }
    // Phase 2: e_user @ u_w  (K = 64..127)
    #pragma unroll
    for (int k = 0; k < DD; k += 4) {
        v2f a = *(const v2f*)(user_emb + uidx * DD + k + 2 * half);
        int kk = k + 2 * half;
        #pragma unroll
        for (int q = 0; q < 4; ++q) {
            v2f bmat;
            bmat.x = u_w[kk * NBI + ccol[q]];
            bmat.y = u_w[(kk + 1) * NBI + ccol[q]];
            acc[q] = __builtin_amdgcn_wmma_f32_16x16x4_f32(
                false, a, false, bmat, (short)0, acc[q], false, false);
        }
    }

    #pragma unroll
    for (int q = 0; q < 4; ++q) {
        int col = nBase + q * 16 + l16;
        if (col < NBI) {
            float ub = u_b[col];
            #pragma unroll
            for (int j = 0; j < 8; ++j) {
                int row = mBase + j + 8 * half;
                __builtin_nontemporal_store(acc[q][j] + ub, &score[row * NBI + col]);
            }
        }
    }
}

// ---------------------------------------------------------------------------
// Kernel 4: sparse old-item corrections.
// score[b,e] += oh_n * (final_each[n,b].e_all[e] - final[b].e_all[e]),
// deduped within each (n,b) row. 4*1024*20 threads, atomic f32 adds.
// ---------------------------------------------------------------------------
__global__ __launch_bounds__(256) void old_correct(
    const int* __restrict__ oldItems,     // (4,1024,20)
    const float* __restrict__ item_emb,
    const float* __restrict__ feo,        // (4,1024,64)
    const float* __restrict__ fin,        // (1024,64)
    float* __restrict__ score)
{
    int t = blockIdx.x * 256 + threadIdx.x;   // < 4*1024*20
    int n = t / (BB * NOLD);
    int r = t - n * (BB * NOLD);
    int b = r / NOLD;
    int j = r - b * NOLD;
    const int* row = oldItems + (n * BB + b) * NOLD;
    int e = row[j];
    bool dup = false;
    for (int j2 = 0; j2 < j; ++j2) dup = dup || (row[j2] == e);
    if (!dup) {
        const float* ea = item_emb + (e + 1) * DD;   // e_all[e]
        const float* fe = feo + (n * BB + b) * DD;
        const float* fi = fin + b * DD;
        float d4 = 0.f, d5 = 0.f;
        #pragma unroll 8
        for (int d = 0; d < DD; ++d) { d4 = fmaf(fe[d], ea[d], d4); d5 = fmaf(fi[d], ea[d], d5); }
        atomicAdd(&score[b * NBI + e], d4 - d5);
    }
}

// ---------------------------------------------------------------------------
// Kernel 5: sim = e_all @ e_all^T via WMMA f32 16x16x4. Store-bandwidth bound.
// Wave computes a 16x96 strip (12000 = 125*96 exact -> no edge clamp, EXEC
// trivially all-1s). A and B lane layouts are symmetric -> both are plain
// float2 loads from row-major e_all. NT stores keep the 576 MB stream out
// of L2.
// ---------------------------------------------------------------------------
#define SIM_NW  (750 * 125)

__global__ __launch_bounds__(256) void sim_gemm(
    const float* __restrict__ item_emb,
    float* __restrict__ sim)              // (12000,12000)
{
    const float* e_all = item_emb + DD;
    const int wid  = blockIdx.x * 8 + (threadIdx.x >> 5);
    if (wid >= SIM_NW) return;            // whole-wave uniform exit
    const int lane = threadIdx.x & 31;
    const int l16  = lane & 15;
    const int half = lane >> 4;
    const int mTile  = wid % 750;         // 750 M tiles exact
    const int nSuper = wid / 750;         // 125 N super-tiles (96 cols each)
    const int mBase = mTile * 16;
    const int nBase = nSuper * 96;
    const int arow  = mBase + l16;

    const v8f vzero = {0.f, 0.f, 0.f, 0.f, 0.f, 0.f, 0.f, 0.f};
    v8f acc[6] = {vzero, vzero, vzero, vzero, vzero, vzero};

    int ccol[6];
    #pragma unroll
    for (int q = 0; q < 6; ++q) ccol[q] = nBase + q * 16 + l16;   // always < 12000

    #pragma unroll
    for (int k = 0; k < DD; k += 4) {
        v2f a = *(const v2f*)(e_all + arow * DD + k + 2 * half);
        #pragma unroll
        for (int q = 0; q < 6; ++q) {
            v2f bmat = *(const v2f*)(e_all + ccol[q] * DD + k + 2 * half);
            acc[q] = __builtin_amdgcn_wmma_f32_16x16x4_f32(
                false, a, false, bmat, (short)0, acc[q], false, false);
        }
    }

    #pragma unroll
    for (int q = 0; q < 6; ++q) {
        int col = ccol[q];
        #pragma unroll
        for (int j = 0; j < 8; ++j) {
            int row = mBase + j + 8 * half;
            __builtin_nontemporal_store(acc[q][j], &sim[(size_t)row * NBI + col]);
        }
    }
}

// ---------------------------------------------------------------------------
// Kernel 6: all_omiga = omiga[1:]
// ---------------------------------------------------------------------------
__global__ __launch_bounds__(256) void omiga_copy(
    const float* __restrict__ omiga, float* __restrict__ out)
{
    int i = blockIdx.x * 256 + threadIdx.x;
    if (i < NBI) out[i] = omiga[i + 1];
}

// ---------------------------------------------------------------------------
extern "C" void kernel_launch(void* const* d_in, const int* in_sizes, int n_in,
                              void* d_out, int out_size, void* d_ws, size_t ws_size,
                              hipStream_t stream)
{
    const int*   clickD   = (const int*)  d_in[0];
    const int*   favorD   = (const int*)  d_in[1];
    const int*   cartD    = (const int*)  d_in[2];
    const int*   buyD     = (const int*)  d_in[3];
    const int*   userD    = (const int*)  d_in[4];
    const int*   oldItems = (const int*)  d_in[5];
    const float* item_emb = (const float*)d_in[6];
    const float* user_emb = (const float*)d_in[7];
    const float* user_wd  = (const float*)d_in[8];
    const float* omiga    = (const float*)d_in[9];
    const float* w_mix    = (const float*)d_in[10];
    const float* alpha    = (const float*)d_in[11];
    const float* gate_w   = (const float*)d_in[12];
    const float* gate_b   = (const float*)d_in[13];
    const float* Wk       = (const float*)d_in[14];
    const float* bk       = (const float*)d_in[15];
    const float* u_w      = (const float*)d_in[16];
    const float* u_b      = (const float*)d_in[17];

    float* out_score = (float*)d_out;                         // (1024,12000)
    float* out_sim   = out_score + (size_t)BB * NBI;          // (12000,12000)
    float* out_omiga = out_sim + (size_t)NBI * NBI;           // (12000,)

    float* ws  = (float*)d_ws;
    float* s1o = ws;                       // 4*1024*64
    float* s2o = s1o + 4 * BB * DD;
    float* s3o = s2o + 4 * BB * DD;
    float* lvo = s3o + 4 * BB * DD;
    float* feo = lvo + 4 * BB * DD;        // final_each (4,1024,64)
    float* fin = feo + 4 * BB * DD;        // final (1024,64)

    // 1. front end: 4096 (n,b) blocks of 64 threads
    front_kernel<<<4 * BB, 64, 0, stream>>>(
        clickD, favorD, cartD, buyD, userD, item_emb, user_emb, user_wd,
        alpha, gate_w, gate_b, Wk, bk, s1o, s2o, s3o, lvo, feo);

    // 2. mix across n -> final
    mix_kernel<<<BB, 64, 0, stream>>>(
        s1o, s2o, s3o, lvo, w_mix, alpha, gate_w, gate_b, fin);

    // 3. dense score GEMM: 64 M-tiles x 188 N-supertiles, 8 waves/block
    score_gemm<<<(64 * 188) / 8, 256, 0, stream>>>(
        fin, user_emb, userD, item_emb, u_w, u_b, out_score);

    // 4. sparse old-item corrections
    old_correct<<<(4 * BB * NOLD) / 256, 256, 0, stream>>>(
        oldItems, item_emb, feo, fin, out_score);

    // 5. sim GEMM: 750 x 125 wave tiles (16x96 each)
    sim_gemm<<<(SIM_NW + 7) / 8, 256, 0, stream>>>(item_emb, out_sim);

    // 6. omiga slice copy
    omiga_copy<<<(NBI + 255) / 256, 256, 0, stream>>>(omiga, out_omiga);
}